// GA_GM_37263136260676
// MI455X (gfx1250) — compile-verified
//
#include <hip/hip_runtime.h>
#include <hip/hip_bf16.h>

// ---------------------------------------------------------------------------
// Graduated-assignment graph matching on MI455X (gfx1250, wave32).
// Core math in fp32 via V_WMMA_F32_16X16X4_F32 (native fp32 WMMA).
// G=16 graphs, M=128, N_UNIV=128, N=2048; 4 taus x 5 GA iters = 20 iterations.
// ---------------------------------------------------------------------------

typedef float v2f __attribute__((ext_vector_type(2)));
typedef float v8f __attribute__((ext_vector_type(8)));

#define NTOT   2048
#define NU     128
#define NGRAPH 16
#define MPERG  128

__device__ __forceinline__ v8f wmma_f32(v2f a, v2f b, v8f c) {
    // 8 args: (neg_a, A, neg_b, B, c_mod, C, reuse_a, reuse_b)
    return __builtin_amdgcn_wmma_f32_16x16x4_f32(
        false, a, false, b, (short)0, c, false, false);
}

// ---------------------------------------------------------------------------
// C[Mx N] = A[M x K] @ B[K x N]  (row-major). One 16x16 tile per wave.
// grid.x = N/16 tile cols, grid.y covers M/(16*8) (8 waves per 256-thr block).
// A-fragment layout (32-bit A 16x4): lanes 0-15 hold K={0,1}, lanes 16-31
// hold K={2,3} -> contiguous float2 load per lane.
// ---------------------------------------------------------------------------
__global__ void __launch_bounds__(256)
gemm_nn(const float* __restrict__ A, int lda,
        const float* __restrict__ B, int ldb,
        float* __restrict__ C, int ldc, int K) {
    const int lane = threadIdx.x & 31;
    const int wave = threadIdx.x >> 5;
    const int half = lane >> 4;
    const int m    = lane & 15;
    const long row0 = (long)(blockIdx.y * 8 + wave) * 16;
    const long col0 = (long)blockIdx.x * 16;

    const float* Ar = A + (row0 + m) * (long)lda;
    const float* Bc = B + col0 + m;

    v8f acc = {};
    #pragma unroll 4
    for (int k = 0; k < K; k += 4) {
        v2f a = *(const v2f*)(Ar + k + 2 * half);        // global_load_b64
        v2f b;
        b.x = Bc[(long)(k + 2 * half)     * ldb];
        b.y = Bc[(long)(k + 2 * half + 1) * ldb];
        acc = wmma_f32(a, b, acc);
    }

    float* Cp = C + col0 + m;
    #pragma unroll
    for (int v = 0; v < 8; ++v)
        Cp[(row0 + v + 8 * half) * (long)ldc] = acc[v];
}

// ---------------------------------------------------------------------------
// Split-K partials of T1 = U^T @ AU.  U, AU: [2048 x 128].
// grid = (8 col tiles, 1, NSPLIT); wave id = row tile (128/16 = 8).
// Deterministic: each split writes its own 128x128 slice.
// ---------------------------------------------------------------------------
__global__ void __launch_bounds__(256)
gemm_tn_splitk(const float* __restrict__ Ua, const float* __restrict__ AU,
               float* __restrict__ part, int kPerSplit) {
    const int lane = threadIdx.x & 31;
    const int wave = threadIdx.x >> 5;
    const int half = lane >> 4;
    const int m    = lane & 15;
    const int tm   = wave;           // row tile of T1
    const int tn   = blockIdx.x;     // col tile of T1
    const int k0   = blockIdx.z * kPerSplit;

    const float* Ap = Ua + tm * 16 + m;   // A^T: a[m][k] = U[k][tm*16+m]
    const float* Bp = AU + tn * 16 + m;

    v8f acc = {};
    #pragma unroll 4
    for (int k = k0; k < k0 + kPerSplit; k += 4) {
        v2f a, b;
        a.x = Ap[(long)(k + 2 * half)     * NU];
        a.y = Ap[(long)(k + 2 * half + 1) * NU];
        b.x = Bp[(long)(k + 2 * half)     * NU];
        b.y = Bp[(long)(k + 2 * half + 1) * NU];
        acc = wmma_f32(a, b, acc);
    }

    float* Cp = part + (long)blockIdx.z * NU * NU + tn * 16 + m;
    #pragma unroll
    for (int v = 0; v < 8; ++v)
        Cp[(long)(tm * 16 + v + 8 * half) * NU] = acc[v];
}

__global__ void reduce_splitk(const float* __restrict__ part,
                              float* __restrict__ T1, int nSplit) {
    int i = blockIdx.x * blockDim.x + threadIdx.x;
    if (i < NU * NU) {
        float s = 0.0f;
        for (int j = 0; j < nSplit; ++j) s += part[(long)j * NU * NU + i];
        T1[i] = s;
    }
}

// ---------------------------------------------------------------------------
// S = scale * (2 * A@T2 + W@U).  A,W: [2048x2048]; T2,U,S: [2048x128].
// Two WMMA accumulator chains per tile; A/W rows share indexing.
// scale = 1/(G*tau) so S is exactly the Sinkhorn log-domain input.
// ---------------------------------------------------------------------------
__global__ void __launch_bounds__(256)
gemm_fused(const float* __restrict__ A, const float* __restrict__ W,
           const float* __restrict__ T2, const float* __restrict__ Uo,
           float* __restrict__ S, float scale) {
    const int lane = threadIdx.x & 31;
    const int wave = threadIdx.x >> 5;
    const int half = lane >> 4;
    const int m    = lane & 15;
    const long row0 = (long)(blockIdx.y * 8 + wave) * 16;
    const long col0 = (long)blockIdx.x * 16;

    const float* Ar = A  + (row0 + m) * (long)NTOT;
    const float* Wr = W  + (row0 + m) * (long)NTOT;
    const float* B1 = T2 + col0 + m;
    const float* B2 = Uo + col0 + m;

    v8f acc1 = {};
    v8f acc2 = {};
    #pragma unroll 2
    for (int k = 0; k < NTOT; k += 4) {
        v2f a = *(const v2f*)(Ar + k + 2 * half);
        v2f w = *(const v2f*)(Wr + k + 2 * half);
        v2f b1, b2;
        b1.x = B1[(long)(k + 2 * half)     * NU];
        b1.y = B1[(long)(k + 2 * half + 1) * NU];
        b2.x = B2[(long)(k + 2 * half)     * NU];
        b2.y = B2[(long)(k + 2 * half + 1) * NU];
        acc1 = wmma_f32(a, b1, acc1);
        acc2 = wmma_f32(w, b2, acc2);
    }

    float* Sp = S + col0 + m;
    #pragma unroll
    for (int v = 0; v < 8; ++v)
        Sp[(row0 + v + 8 * half) * (long)NU] = (2.0f * acc1[v] + acc2[v]) * scale;
}

// ---------------------------------------------------------------------------
// Log-domain Sinkhorn, 20 alternating row/col logsumexp passes per graph.
// One block (128 threads, 4 waves) per graph; 128x128 fp32 tile = 64KB LDS.
// Row pass skews the column order by lane id -> conflict-free LDS banks.
// ---------------------------------------------------------------------------
__global__ void __launch_bounds__(128)
sinkhorn_kernel(const float* __restrict__ Vs, float* __restrict__ U) {
    __shared__ float s[MPERG * NU];        // exactly 64KB
    const int g   = blockIdx.x;
    const int tid = threadIdx.x;           // 0..127
    const float* src = Vs + (long)g * MPERG * NU;
    for (int i = tid; i < MPERG * NU; i += 128) s[i] = src[i];
    __syncthreads();

    for (int it = 0; it < 20; ++it) {
        if ((it & 1) == 0) {
            // axis=2: normalize row `tid` over its 128 columns (skewed order)
            const int r = tid;
            float mx = -3.402823466e38f;
            for (int i = 0; i < NU; ++i) {
                int c = (r + i) & (NU - 1);
                mx = fmaxf(mx, s[r * NU + c]);
            }
            float sum = 0.0f;
            for (int i = 0; i < NU; ++i) {
                int c = (r + i) & (NU - 1);
                sum += expf(s[r * NU + c] - mx);
            }
            const float lse = mx + logf(sum);
            for (int i = 0; i < NU; ++i) {
                int c = (r + i) & (NU - 1);
                s[r * NU + c] -= lse;
            }
        } else {
            // axis=1: normalize column `tid` over the 128 rows of this graph
            const int c = tid;
            float mx = -3.402823466e38f;
            for (int r = 0; r < MPERG; ++r) mx = fmaxf(mx, s[r * NU + c]);
            float sum = 0.0f;
            for (int r = 0; r < MPERG; ++r) sum += expf(s[r * NU + c] - mx);
            const float lse = mx + logf(sum);
            for (int r = 0; r < MPERG; ++r) s[r * NU + c] -= lse;
        }
        __syncthreads();
    }

    float* dst = U + (long)g * MPERG * NU;
    for (int i = tid; i < MPERG * NU; i += 128) dst[i] = expf(s[i]);
}

// Copy U to d_out and zero the 16 int32 cluster slots (bit pattern 0).
__global__ void finalize_kernel(const float* __restrict__ U, float* __restrict__ out) {
    int i = blockIdx.x * blockDim.x + threadIdx.x;
    if (i < NTOT * NU)            out[i] = U[i];
    else if (i < NTOT * NU + 16)  out[i] = 0.0f;
}

// ---------------------------------------------------------------------------
extern "C" void kernel_launch(void* const* d_in, const int* in_sizes, int n_in,
                              void* d_out, int out_size, void* d_ws, size_t ws_size,
                              hipStream_t stream) {
    (void)in_sizes; (void)n_in; (void)out_size; (void)ws_size;
    const float* A  = (const float*)d_in[0];
    const float* W  = (const float*)d_in[1];
    const float* U0 = (const float*)d_in[2];
    float* out = (float*)d_out;

    // Workspace layout (floats): ~5.1 MB total
    float* U    = (float*)d_ws;                  // 2048*128
    float* AU   = U    + NTOT * NU;              // 2048*128
    float* T1   = AU   + NTOT * NU;              // 128*128
    float* T2   = T1   + NU * NU;                // 2048*128
    float* Vs   = T2   + NTOT * NU;              // 2048*128
    float* part = Vs   + NTOT * NU;              // 16 * 128*128 split-K partials

    hipMemcpyAsync(U, U0, (size_t)NTOT * NU * sizeof(float),
                   hipMemcpyDeviceToDevice, stream);

    const float taus[4] = {0.5f, 0.25f, 0.125f, 0.0625f};
    const dim3 gemmGrid(NU / 16, NTOT / (16 * 8));     // (8, 16)
    const int  NSPLIT = 16;
    const int  kPerSplit = NTOT / NSPLIT;              // 128

    for (int t = 0; t < 4; ++t) {
        const float scale = 1.0f / ((float)NGRAPH * taus[t]);
        for (int it = 0; it < 5; ++it) {
            // AU = A @ U
            gemm_nn<<<gemmGrid, 256, 0, stream>>>(A, NTOT, U, NU, AU, NU, NTOT);
            // T1 = U^T @ AU  (split-K, deterministic reduce)
            gemm_tn_splitk<<<dim3(NU / 16, 1, NSPLIT), 256, 0, stream>>>(
                U, AU, part, kPerSplit);
            reduce_splitk<<<(NU * NU + 255) / 256, 256, 0, stream>>>(part, T1, NSPLIT);
            // T2 = U @ T1
            gemm_nn<<<gemmGrid, 256, 0, stream>>>(U, NU, T1, NU, T2, NU, NU);
            // Vs = (2*A@T2 + W@U) / (G*tau)
            gemm_fused<<<gemmGrid, 256, 0, stream>>>(A, W, T2, U, Vs, scale);
            // U = exp(sinkhorn(Vs))
            sinkhorn_kernel<<<NGRAPH, 128, 0, stream>>>(Vs, U);
        }
    }

    finalize_kernel<<<(NTOT * NU + 16 + 255) / 256, 256, 0, stream>>>(U, out);
}